// CRF_1992864825723
// MI455X (gfx1250) — compile-verified
//
#include <hip/hip_runtime.h>

// CRF Viterbi decode for MI455X (gfx1250, wave32).
//
// feats: [S=2048, B=4096, T=10] f32; transitions[next, prev]: [10,10] f32.
// Outputs (concatenated in d_out as float): path_score [B], best_path [S, B].
//
// Forward: 10 lanes per batch element (3 groups / wave32), ds_bpermute fv
// gather, 8-deep GLOBAL_LOAD_ASYNC_TO_LDS_B32 feature staging (ASYNCcnt,
// per-wave buffers -> no barriers). Backtrack: associative composition of
// the per-step [10]->[10] backpointer maps (nibble-packed u64 per (s,b)).
//
// Workspace layout (assumed ws_size >= ~68 MB):
//   bp64     : u64[S*B]      (64 MB)  10 nibbles = backpointers per (s,b)
//   Fm       : u64[NC*B]     ( 2 MB)  per-chunk composed maps
//   carry    : i32[NC*B]     ( 1 MB)  tag entering each chunk (from the top)
//   bestlast : i32[B]        (16 KB)  argmax of terminal scores

#define SEQ_LEN   2048
#define BATCH     4096
#define TAGS      10
#define START_TAG 8
#define STOP_TAG  9
#define NEG_INF   (-10000.0f)
#define CHUNK     32
#define NCHUNK    (SEQ_LEN / CHUNK)   // 64

#define FWD_BLOCK 256
#define DEPTH     8                    // async pipeline depth (steps in flight)
#define NSLOT     16                   // LDS slots (2x depth -> no reuse hazard)

__device__ __forceinline__ float bperm_f(int srcLane, float v) {
    return __int_as_float(__builtin_amdgcn_ds_bpermute(srcLane << 2, __float_as_int(v)));
}
__device__ __forceinline__ int bperm_i(int srcLane, int v) {
    return __builtin_amdgcn_ds_bpermute(srcLane << 2, v);
}

// Fire-and-forget 4B load into LDS, tracked by ASYNCcnt (gfx1250).
// lds_addr = low 32 bits of the flat pointer (LDS aperture: addr[31:0]).
__device__ __forceinline__ void async_load_f32(unsigned lds_addr, const float* gptr) {
    unsigned long long ga = (unsigned long long)gptr;
    asm volatile("global_load_async_to_lds_b32 %0, %1, off"
                 :: "v"(lds_addr), "v"(ga)
                 : "memory");
}

// ---------------------------------------------------------------------------
// Forward pass: 10 lanes per batch element, 3 groups per wave32.
// Lane layout within a wave: lane = g*10 + tag (g in 0..2; lanes 30,31 idle).
// ---------------------------------------------------------------------------
__global__ __launch_bounds__(FWD_BLOCK) void crf_forward(
    const float* __restrict__ feats,
    const float* __restrict__ trans,
    float* __restrict__ path_score,
    int*   __restrict__ bestlast,
    unsigned char* __restrict__ bp_bytes)  // u64-strided, low 5 bytes per (s,b)
{
    __shared__ float stage[NSLOT][FWD_BLOCK];   // 16 KB: per-wave async staging

    const int tid   = threadIdx.x;
    const int lane  = tid & 31;
    const int wid   = tid >> 5;
    const int gin   = lane / 10;            // 0..2 active, 3 => idle lane
    const int tag   = lane - gin * 10;      // this lane's "next" tag
    const int gbase = gin * 10;
    const int b     = (blockIdx.x * (FWD_BLOCK / 32) + wid) * 3 + gin;
    const bool active = (gin < 3) && (b < BATCH);

    // Per-lane transition row trans[tag][p] (10 VGPRs) + terminal column entry.
    float tr[TAGS];
    float trStop = 0.0f;
    if (active) {
#pragma unroll
        for (int p = 0; p < TAGS; ++p) tr[p] = trans[tag * TAGS + p];
        trStop = trans[STOP_TAG * TAGS + tag];
    }

    float fv = (tag == START_TAG) ? 0.0f : NEG_INF;

    const size_t fstep = (size_t)BATCH * TAGS;           // elements per step
    const float* fp0   = feats + (size_t)b * TAGS + tag; // lane's step-0 elem
    const unsigned ldsBase = (unsigned)(unsigned long long)(const void*)&stage[0][tid];

    // Prologue: put DEPTH steps in flight.
    if (active) {
#pragma unroll
        for (int k = 0; k < DEPTH; ++k)
            async_load_f32(ldsBase + (unsigned)(k * FWD_BLOCK * 4),
                           fp0 + (size_t)k * fstep);
    }

    // One forward step: gather 10 prev fv via ds_bpermute, max/argmax with
    // strict '>' (first max wins, matching jnp.argmax), add feature, pack bp.
    auto step_body = [&](int s, float fcur) {
        float best = bperm_f(gbase + 0, fv) + tr[0];
        int   bp   = 0;
#pragma unroll
        for (int p = 1; p < TAGS; ++p) {
            float c = bperm_f(gbase + p, fv) + tr[p];
            if (c > best) { best = c; bp = p; }
        }
        fv = best + fcur;

        // Nibble of tag n lives at bit 4n of the (s,b) u64.
        // Even-tag lanes store one byte = own nibble | partner nibble << 4.
        int bpHi = bperm_i(lane + 1, bp);
        if (active && ((tag & 1) == 0)) {
            bp_bytes[((size_t)s * BATCH + b) * 8 + (tag >> 1)] =
                (unsigned char)(bp | (bpHi << 4));
        }
    };

    int s = 0;
    // Steady state: 8 in flight; wait <=7 so the oldest (step s) has landed.
    for (; s < SEQ_LEN - DEPTH; ++s) {
        asm volatile("s_wait_asynccnt 0x7" ::: "memory");
        float fcur = stage[s & (NSLOT - 1)][tid];
        if (active) {
            const int sn = s + DEPTH;
            async_load_f32(ldsBase + (unsigned)((sn & (NSLOT - 1)) * FWD_BLOCK * 4),
                           fp0 + (size_t)sn * fstep);
            if (sn + DEPTH < SEQ_LEN)   // warm L2 another DEPTH ahead
                __builtin_prefetch(fp0 + (size_t)(sn + DEPTH) * fstep, 0, 1);
        }
        step_body(s, fcur);
    }
    // Tail: fewer than DEPTH outstanding -> wait for everything.
    for (; s < SEQ_LEN; ++s) {
        asm volatile("s_wait_asynccnt 0x0" ::: "memory");
        float fcur = stage[s & (NSLOT - 1)][tid];
        step_body(s, fcur);
    }

    // Terminal: term[p] = fv[p] + trans[STOP][p]; group max/argmax.
    float term = fv + trStop;
    float best = bperm_f(gbase + 0, term);
    int   arg  = 0;
#pragma unroll
    for (int p = 1; p < TAGS; ++p) {
        float c = bperm_f(gbase + p, term);
        if (c > best) { best = c; arg = p; }
    }
    if (active && tag == 0) {
        path_score[b] = best;
        bestlast[b]   = arg;
    }
}

// ---------------------------------------------------------------------------
// Phase A: compose each chunk's 32 backpointer maps into one [10]->[10] map.
// Folding top-down: m_new[x] = bp[t][ m[x] ].  All loads address-independent.
// ---------------------------------------------------------------------------
__global__ __launch_bounds__(256) void crf_chunk_maps(
    const unsigned long long* __restrict__ bp64,
    unsigned long long* __restrict__ Fm)
{
    const int idx = blockIdx.x * blockDim.x + threadIdx.x;
    const int b = idx & (BATCH - 1);
    const int c = idx >> 12;                 // idx / 4096
    if (c >= NCHUNK) return;

    unsigned long long m = 0x9876543210ULL;  // identity map (nibbles 0..9)
    const int tTop = c * CHUNK + CHUNK - 1;
    for (int k = 0; k < CHUNK; ++k) {
        const unsigned long long w = bp64[(size_t)(tTop - k) * BATCH + b];
        unsigned long long nm = 0ULL;
#pragma unroll
        for (int x = 0; x < TAGS; ++x) {
            const unsigned d = (unsigned)(m >> (4 * x)) & 15u;    // current carry
            const unsigned long long e = (w >> (4u * d)) & 15ULL; // bp[t][d]
            nm |= e << (4 * x);
        }
        m = nm;
    }
    Fm[(size_t)c * BATCH + b] = m;
}

// ---------------------------------------------------------------------------
// Phase B: chain the 64 chunk maps per batch element (cheap, 4096 threads).
// carry[c] = tag at the TOP of chunk c (i.e., the emission at t = c*L + L-1).
// ---------------------------------------------------------------------------
__global__ __launch_bounds__(256) void crf_boundary(
    const unsigned long long* __restrict__ Fm,
    const int* __restrict__ bestlast,
    int* __restrict__ carry)
{
    const int b = blockIdx.x * blockDim.x + threadIdx.x;
    if (b >= BATCH) return;
    unsigned cur = (unsigned)bestlast[b];
    for (int c = NCHUNK - 1; c >= 0; --c) {
        carry[c * BATCH + b] = (int)cur;
        cur = (unsigned)(Fm[(size_t)c * BATCH + b] >> (4u * cur)) & 15u;
    }
}

// ---------------------------------------------------------------------------
// Phase C: emit the decoded path within each chunk in parallel.
// best_path[t] = carry BEFORE applying bp[t] (matches the reference scan).
// ---------------------------------------------------------------------------
__global__ __launch_bounds__(256) void crf_emit(
    const unsigned long long* __restrict__ bp64,
    const int* __restrict__ carry,
    float* __restrict__ best_path)
{
    const int idx = blockIdx.x * blockDim.x + threadIdx.x;
    const int b = idx & (BATCH - 1);
    const int c = idx >> 12;
    if (c >= NCHUNK) return;

    unsigned cur = (unsigned)carry[c * BATCH + b];
    const int tTop = c * CHUNK + CHUNK - 1;
    for (int k = 0; k < CHUNK; ++k) {
        const int t = tTop - k;
        best_path[(size_t)t * BATCH + b] = (float)cur;
        cur = (unsigned)(bp64[(size_t)t * BATCH + b] >> (4u * cur)) & 15u;
    }
}

// ---------------------------------------------------------------------------
extern "C" void kernel_launch(void* const* d_in, const int* in_sizes, int n_in,
                              void* d_out, int out_size, void* d_ws, size_t ws_size,
                              hipStream_t stream)
{
    (void)in_sizes; (void)n_in; (void)out_size; (void)ws_size;

    const float* feats = (const float*)d_in[0];
    const float* trans = (const float*)d_in[1];

    float* path_score = (float*)d_out;            // [B]
    float* best_path  = (float*)d_out + BATCH;    // [S, B]

    char* ws = (char*)d_ws;
    unsigned long long* bp64 = (unsigned long long*)ws;                       // 64 MB
    size_t off = (size_t)SEQ_LEN * BATCH * 8;
    unsigned long long* Fm = (unsigned long long*)(ws + off);                 //  2 MB
    off += (size_t)NCHUNK * BATCH * 8;
    int* carry = (int*)(ws + off);                                            //  1 MB
    off += (size_t)NCHUNK * BATCH * 4;
    int* bestlast = (int*)(ws + off);                                         // 16 KB

    const dim3 blk(256);
    const int fwdBlocks = (BATCH + 23) / 24;      // 3 groups/wave, 8 waves/block

    hipLaunchKernelGGL(crf_forward, dim3(fwdBlocks), blk, 0, stream,
                       feats, trans, path_score, bestlast, (unsigned char*)bp64);
    hipLaunchKernelGGL(crf_chunk_maps, dim3((NCHUNK * BATCH) / 256), blk, 0, stream,
                       bp64, Fm);
    hipLaunchKernelGGL(crf_boundary, dim3(BATCH / 256), blk, 0, stream,
                       Fm, bestlast, carry);
    hipLaunchKernelGGL(crf_emit, dim3((NCHUNK * BATCH) / 256), blk, 0, stream,
                       bp64, carry, best_path);
}